// NonMaximaSuppression3d_43267500540175
// MI455X (gfx1250) — compile-verified
//
#include <hip/hip_runtime.h>

// ---------------------------------------------------------------------------
// 3D NMS (26-neighbor strict max) on (B*C, 128, 256, 256) fp32.
// Bandwidth-bound: 128 MiB in + 128 MiB out => ~11.5 us floor @ 23.3 TB/s.
// CDNA5 path: TENSOR_LOAD_TO_LDS (TDM DMA) stages a haloed 3D tile in LDS,
// waves compute rolling 3x3x3 max from LDS, write once with NT stores.
// ---------------------------------------------------------------------------

#define TW 256   // tensor W (fastest)
#define TH 256   // tensor H
#define TD 128   // tensor D
#define WT 64    // tile W (outputs)
#define HT 8     // tile H
#define DT 8     // tile D
#define SMEM_ELEMS ((DT + 2) * (HT + 2) * (WT + 2))   // 10*10*66 = 6600 floats

typedef __attribute__((ext_vector_type(4))) unsigned int u32x4;
typedef __attribute__((ext_vector_type(8))) int          i32x8;
typedef __attribute__((ext_vector_type(4))) int          i32x4;

// Build a 3D Tensor-DMA descriptor (D#) and issue TENSOR_LOAD_TO_LDS.
// Tensor: dim0=TW (stride 1), dim1=TH (stride TW), dim2=TD (stride TW*TH).
// Tile:   LW x LH x LD elements starting at gptr; always fully in-bounds
//         (caller clamps), packed row-major into LDS at lds_addr.
// This toolchain's builtin takes 6 args: G0(v4u), G1(v8i), G2(v4i), G3(v4i),
// an extra 256-bit group (zero-filled / unused here), and cpol.
__device__ __forceinline__ void tdm_load_tile3d(unsigned lds_addr,
                                                const float* gptr,
                                                int LW, int LH, int LD) {
  unsigned long long ga = (unsigned long long)(uintptr_t)gptr;

  u32x4 g0;
  g0[0] = 1u;                                   // count=1 (valid), user mode
  g0[1] = lds_addr;                             // LDS byte address
  g0[2] = (unsigned)ga;                         // global_addr[31:0]
  g0[3] = (unsigned)((ga >> 32) & 0x01FFFFFFu)  // global_addr[56:32]
        | 0x80000000u;                          // type=2 ("image")

  i32x8 g1;
  g1[0] = 0x00020000;                               // data_size=2 (4 bytes)
  g1[1] = (int)(((unsigned)TW & 0xFFFFu) << 16);    // tensor_dim0 lo16
  g1[2] = (int)(((unsigned)TW >> 16) |
                (((unsigned)TH & 0xFFFFu) << 16));  // dim0 hi | dim1 lo
  g1[3] = (int)(((unsigned)TH >> 16) |
                ((unsigned)LW << 16));              // dim1 hi | tile_dim0
  g1[4] = (int)((unsigned)LH | ((unsigned)LD << 16)); // tile_dim1 | tile_dim2
  g1[5] = TW;                                       // dim0_stride[31:0]
  g1[6] = (int)((((unsigned)(TW * TH)) & 0xFFFFu) << 16); // d0s hi16 | d1s lo16
  g1[7] = (int)((unsigned)((unsigned long long)(TW * TH) >> 16)); // d1s[47:16]

  i32x4 g2;
  g2[0] = TD;   // tensor_dim2
  g2[1] = 1;    // tensor_dim3 (unused; tile_dim3=0)
  g2[2] = 0;    // tensor_dim2_stride lo (unused for 3D tiles)
  g2[3] = 0;    // stride hi | tile_dim3=0

  i32x4 g3 = {0, 0, 0, 0};
  i32x8 g4 = {0, 0, 0, 0, 0, 0, 0, 0};   // extra group: zero-filled

  __builtin_amdgcn_tensor_load_to_lds(g0, g1, g2, g3, g4, 0);
}

__global__ __launch_bounds__(256) void nms3d_kernel(const float* __restrict__ x,
                                                    float* __restrict__ out) {
  __shared__ float smem[SMEM_ELEMS];

  const int ztiles = TD / DT;                 // 16
  const int xt = blockIdx.x;                  // 0..TW/WT-1
  const int yt = blockIdx.y;                  // 0..TH/HT-1
  const int zt = blockIdx.z % ztiles;
  const int bc = blockIdx.z / ztiles;

  const int x0 = xt * WT, y0 = yt * HT, z0 = zt * DT;
  // Haloed, clamped-to-tensor load region per dim.
  const int gx = x0 ? x0 - 1 : 0;
  const int gy = y0 ? y0 - 1 : 0;
  const int gz = z0 ? z0 - 1 : 0;
  const int LW = min(x0 + WT + 1, TW) - gx;   // <= WT+2
  const int LH = min(y0 + HT + 1, TH) - gy;   // <= HT+2
  const int LD = min(z0 + DT + 1, TD) - gz;   // <= DT+2

  const float* gbase =
      x + (((size_t)bc * TD + gz) * TH + gy) * (size_t)TW + gx;

  // One wave issues the tensor DMA; done-signal is per issuing wave, so it
  // waits TENSORcnt==0, then the workgroup barrier publishes the LDS tile.
  if (threadIdx.x < 32) {
    // local->flat addrspacecast puts the LDS byte offset in the low 32 bits.
    unsigned lds_addr = (unsigned)(uintptr_t)(void*)&smem[0];
    tdm_load_tile3d(lds_addr, gbase, LW, LH, LD);
    __builtin_amdgcn_s_wait_tensorcnt(0);
  }
  __syncthreads();

  const int tid = threadIdx.x;
  const int lwx = tid & (WT - 1);             // 0..63  (lane-contiguous in W)
  const int hq  = tid >> 6;                   // 0..3
  const int ox = x0 - gx;                     // 0 or 1: output origin in LDS
  const int oy = y0 - gy;
  const int oz = z0 - gz;

  const int  w   = x0 + lwx;
  const bool wok = (w >= 1) & (w <= TW - 2);
  // Clamped LDS center coords: only non-interior voxels (discarded anyway)
  // get clamped, so results are unaffected and accesses stay in-array.
  const int lwc = max(1, min(lwx + ox, LW - 2));
  const int planeStride = LH * LW;

  for (int col = 0; col < 2; ++col) {
    const int  hh   = hq + col * 4;           // 0..7
    const int  h    = y0 + hh;
    const bool hwok = wok & (h >= 1) & (h <= TH - 2);
    const int  lhc  = max(1, min(hh + oy, LH - 2));
    const float* base = smem + (lhc - 1) * LW + (lwc - 1);

    // 3x3 in-plane stats: c = center, m8 = max of 8 in-plane neighbors.
    auto plane_stats = [&](int q, float& c, float& m8) {
      const int qc = max(0, min(q, LD - 1));
      const float* p0 = base + qc * planeStride;
      const float* p1 = p0 + LW;
      const float* p2 = p1 + LW;
      float a0 = p0[0], a1 = p0[1], a2 = p0[2];
      float b0 = p1[0], b1 = p1[1], b2 = p1[2];
      float d0 = p2[0], d1 = p2[1], d2 = p2[2];
      c  = b1;
      m8 = fmaxf(fmaxf(fmaxf(a0, a1), fmaxf(a2, b0)),
                 fmaxf(fmaxf(b2, d0), fmaxf(d1, d2)));
    };

    // Rolling window over depth: voxel dz needs planes oz+dz-1 .. oz+dz+1.
    float cP, m8P, m9_below;
    {
      float cB, m8B;
      plane_stats(oz - 1, cB, m8B);
      m9_below = fmaxf(m8B, cB);
      plane_stats(oz, cP, m8P);
    }

    size_t obase = (((size_t)bc * TD + z0) * TH + h) * (size_t)TW + w;
    for (int dz = 0; dz < DT; ++dz) {
      float cN, m8N;
      plane_stats(oz + dz + 1, cN, m8N);
      const float m9_above = fmaxf(m8N, cN);

      const int  d  = z0 + dz;
      const bool ok = hwok & (d >= 1) & (d <= TD - 2) &
                      (cP > fmaxf(fmaxf(m9_below, m8P), m9_above));
      const float res = ok ? cP : 0.0f;
      __builtin_nontemporal_store(res, &out[obase + (size_t)dz * (TH * TW)]);

      m9_below = fmaxf(m8P, cP);
      cP = cN;
      m8P = m8N;
    }
  }
}

extern "C" void kernel_launch(void* const* d_in, const int* in_sizes, int n_in,
                              void* d_out, int out_size, void* d_ws,
                              size_t ws_size, hipStream_t stream) {
  (void)n_in; (void)out_size; (void)d_ws; (void)ws_size;
  const float* xin = (const float*)d_in[0];
  float* outp = (float*)d_out;
  const int nBC = in_sizes[0] / (TD * TH * TW);   // = 4 for (2,2,128,256,256)
  dim3 grid(TW / WT, TH / HT, (TD / DT) * nBC);   // 4 x 32 x 64
  nms3d_kernel<<<grid, 256, 0, stream>>>(xin, outp);
}